// GraspCVAELoss_80006650790046
// MI455X (gfx1250) — compile-verified
//
#include <hip/hip_runtime.h>
#include <hip/hip_bf16.h>

#define NB 32
#define NO 3000
#define NH 778
#define NZ 64
#define NPRIOR 204
#define NTILES ((NH + 15) / 16)   // 49 column tiles
#define ROWS_PER_BLOCK 128        // 8 waves * 16 rows
#define BIGF 3.0e38f

typedef __attribute__((ext_vector_type(2))) float v2f;
typedef __attribute__((ext_vector_type(8))) float v8f;

__constant__ int c_prior[NPRIOR] = {
  697, 698, 699, 700, 712, 713, 714, 715, 737, 738, 739, 740, 741, 743, 744, 745,
  746, 748, 749, 750, 753, 754, 755, 756, 757, 758, 759, 760, 761, 762, 763, 764,
  765, 766, 767, 768,
  46, 47, 48, 49, 164, 165, 166, 167, 194, 195, 223, 237, 238, 280, 281, 298,
  301, 317, 320, 323, 324, 325, 326, 327, 328, 329, 330, 331, 332, 333, 340, 341,
  342, 343, 344, 345, 346, 347, 348, 349, 350, 351, 352, 353, 354, 355,
  356, 357, 358, 359, 375, 376, 386, 387, 396, 397, 402, 403, 413, 429, 433, 434,
  435, 436, 437, 438, 439, 440, 441, 442, 443, 444, 452, 453, 454, 455, 456, 459,
  460, 461, 462, 463, 464, 465, 466, 467,
  468, 469, 470, 471, 484, 485, 486, 496, 497, 506, 507, 513, 514, 524, 545, 546,
  547, 548, 549, 550, 551, 552, 553, 555, 563, 564, 565, 566, 567, 570, 572, 573,
  574, 575, 576, 577, 578,
  580, 581, 582, 583, 600, 601, 602, 614, 615, 624, 625, 630, 631, 641, 663, 664,
  665, 666, 667, 668, 670, 672, 680, 681, 682, 683, 684, 686, 687, 688, 689, 690,
  691, 692, 693, 694, 695,
  73, 96, 98, 99, 772, 774, 775, 777
};

__device__ __forceinline__ float sgnf(float x) {
  return (x > 0.f) ? 1.f : ((x < 0.f) ? -1.f : 0.f);
}

// XOR-swizzle cross-lane exchange: one ds_swizzle_b32, immediate pattern,
// no address VGPR (group-of-32 mode: offset[14:10]=xor, [4:0]=and).
template <int M>
__device__ __forceinline__ float swz_f(float x) {
  return __int_as_float(
      __builtin_amdgcn_ds_swizzle(__float_as_int(x), (M << 10) | 0x1f));
}
template <int M>
__device__ __forceinline__ int swz_i(int x) {
  return __builtin_amdgcn_ds_swizzle(x, (M << 10) | 0x1f);
}

__device__ __forceinline__ float wave_sum(float v) {
  v += swz_f<1>(v);
  v += swz_f<2>(v);
  v += swz_f<4>(v);
  v += swz_f<8>(v);
  v += swz_f<16>(v);
  return v;
}

template <int M>
__device__ __forceinline__ void red_step(float& rm, int& ri, float& pm) {
  float orm = swz_f<M>(rm);
  int   ori = swz_i<M>(ri);
  float opm = swz_f<M>(pm);
  if (orm < rm) { rm = orm; ri = ori; }
  pm = fminf(pm, opm);
}

// ---------------------------------------------------------------------------
// Init: zero accumulators, set hand->obj min arrays to +inf bit pattern.
// ---------------------------------------------------------------------------
__global__ void gcl_init(unsigned* __restrict__ h2o_min,
                         unsigned* __restrict__ h2ogt_min,
                         float* __restrict__ acc) {
  int t = blockIdx.x * blockDim.x + threadIdx.x;
  if (t < 16) acc[t] = 0.f;
  if (t < NB * NH) {
    h2o_min[t]   = 0x7F800000u;
    h2ogt_min[t] = 0x7F800000u;
  }
}

// ---------------------------------------------------------------------------
// Distance kernel: per (row-chunk, batch, pass) block.
// pass 0: hand = recon (also produces prior-restricted min), pass 1: hand = gt
// Row-min+argmin (obj->hand) via WMMA tiles + swizzle butterfly;
// col-min (hand->obj) via LDS atomic min on f32 bit patterns.
// One V_WMMA_F32_16X16X4_F32 per 16x16 tile:
//   A[m] = (-2ox,-2oy,-2oz, 1), B[n] = (hx,hy,hz,|h|^2) -> -2 o.h + |h|^2
// ---------------------------------------------------------------------------
__global__ __launch_bounds__(256) void gcl_dist(
    const float* __restrict__ obj,    // [B,NO,3]
    const float* __restrict__ recon,  // [B,NH,3]
    const float* __restrict__ gtp,    // [B,NH,3]
    float* __restrict__ recon_d2, int* __restrict__ recon_idx,
    float* __restrict__ gt_d2,    int* __restrict__ gt_idx,
    float* __restrict__ prior_d2,
    unsigned* __restrict__ h2o_min, unsigned* __restrict__ h2ogt_min) {
  __shared__ float4   sHand[NH];
  __shared__ float    sPen[NH];     // 0 if prior column, BIGF otherwise
  __shared__ unsigned sColMin[NH];

  const int b    = blockIdx.y;
  const int pass = blockIdx.z;
  const int tid  = threadIdx.x;
  const float* hand = pass ? gtp : recon;

  for (int i = tid; i < NH; i += 256) {
    float x = hand[(b * NH + i) * 3 + 0];
    float y = hand[(b * NH + i) * 3 + 1];
    float z = hand[(b * NH + i) * 3 + 2];
    sHand[i]   = make_float4(x, y, z, x * x + y * y + z * z);
    sPen[i]    = BIGF;
    sColMin[i] = 0x7F800000u;
  }
  __syncthreads();
  if (tid < NPRIOR) sPen[c_prior[tid]] = 0.f;
  __syncthreads();

  const int wave = tid >> 5;
  const int lane = tid & 31;
  const int hi   = lane >> 4;   // 0: lanes 0-15, 1: lanes 16-31
  const int l16  = lane & 15;
  const int rb   = blockIdx.x * ROWS_PER_BLOCK + wave * 16;

  // A operand: row rb + l16, K split across lane halves.
  float ox = 1.0e6f, oy = 1.0e6f, oz = 1.0e6f;
  {
    int r = rb + l16;
    if (r < NO) {
      ox = obj[(b * NO + r) * 3 + 0];
      oy = obj[(b * NO + r) * 3 + 1];
      oz = obj[(b * NO + r) * 3 + 2];
    }
  }
  v2f A;
  A.x = hi ? (-2.f * oz) : (-2.f * ox);
  A.y = hi ? 1.0f        : (-2.f * oy);

  // Row norms for this lane's 8 C-matrix rows (M = v + 8*hi).
  float onorm[8];
#pragma unroll
  for (int v = 0; v < 8; ++v) {
    int r = rb + hi * 8 + v;
    if (r < NO) {
      float x = obj[(b * NO + r) * 3 + 0];
      float y = obj[(b * NO + r) * 3 + 1];
      float z = obj[(b * NO + r) * 3 + 2];
      onorm[v] = x * x + y * y + z * z;
    } else {
      onorm[v] = 3.0e12f;
    }
  }

  float rmin[8], pmin[8];
  int   ridx[8];
#pragma unroll
  for (int v = 0; v < 8; ++v) { rmin[v] = BIGF; pmin[v] = BIGF; ridx[v] = 0; }

  // Software pipeline: prefetch tile ct+1 while processing tile ct.
  float4 hc = sHand[l16];   // tile 0 (always valid: l16 < 16 <= NH)
  float  pc = sPen[l16];

  for (int ct = 0; ct < NTILES; ++ct) {
    int  col = ct * 16 + l16;
    bool cv  = col < NH;

    // Prefetch next tile (clamped index; junk on last iter, never used).
    int nidx = (col + 16 < NH) ? (col + 16) : 0;
    float4 hn = sHand[nidx];
    float  pn = sPen[nidx];

    float4 h = hc;
    float  pen = cv ? pc : BIGF;
    if (!cv) { h.x = 1.0e6f; h.y = 1.0e6f; h.z = 1.0e6f; h.w = 3.0e12f; }

    v2f Bv;
    Bv.x = hi ? h.z : h.x;
    Bv.y = hi ? h.w : h.y;

    v8f C = {};
    C = __builtin_amdgcn_wmma_f32_16x16x4_f32(
        /*neg_a=*/false, A, /*neg_b=*/false, Bv,
        /*c_mod=*/(short)0, C, /*reuse_a=*/false, /*reuse_b=*/false);

    float lmin = BIGF;
#pragma unroll
    for (int v = 0; v < 8; ++v) {
      float d2 = fmaxf(C[v] + onorm[v], 0.f);
      lmin = fminf(lmin, d2);
      if (d2 < rmin[v]) { rmin[v] = d2; ridx[v] = col; }
      pmin[v] = fminf(pmin[v], d2 + pen);
    }
    // Combine lane/lane^16 partners (same column), one atomic per column.
    lmin = fminf(lmin, swz_f<16>(lmin));
    if (cv && hi == 0) atomicMin(&sColMin[col], __float_as_uint(lmin));

    hc = hn; pc = pn;
  }

  // Cross-lane row-min reduction within each 16-lane half (xor < 16).
#pragma unroll
  for (int v = 0; v < 8; ++v) {
    float rm = rmin[v]; int ri = ridx[v]; float pm = pmin[v];
    red_step<1>(rm, ri, pm);
    red_step<2>(rm, ri, pm);
    red_step<4>(rm, ri, pm);
    red_step<8>(rm, ri, pm);
    if (l16 == 0) {
      int r = rb + hi * 8 + v;
      if (r < NO) {
        int o = b * NO + r;
        if (pass == 0) {
          recon_d2[o] = rm; recon_idx[o] = ri; prior_d2[o] = pm;
        } else {
          gt_d2[o] = rm; gt_idx[o] = ri;
        }
      }
    }
  }

  __syncthreads();
  unsigned* g = pass ? h2ogt_min : h2o_min;
  for (int i = tid; i < NH; i += 256) atomicMin(&g[b * NH + i], sColMin[i]);
}

// ---------------------------------------------------------------------------
// Per obj-point terms: penetration, contact, consistency, loss_o.
// acc: [0]=sum_pen [1]=sum_contact [2]=n_pts [3]=sum_consist [4]=sum_loss_o
// ---------------------------------------------------------------------------
__global__ __launch_bounds__(256) void gcl_post_obj(
    const float* __restrict__ obj,
    const float* __restrict__ recon, const float* __restrict__ rnorm,
    const float* __restrict__ gtp,   const float* __restrict__ gnorm,
    const float* __restrict__ recon_d2, const int* __restrict__ recon_idx,
    const float* __restrict__ gt_d2,    const int* __restrict__ gt_idx,
    const float* __restrict__ prior_d2,
    float* __restrict__ acc) {
  int j = blockIdx.x * blockDim.x + threadIdx.x;
  int b = blockIdx.y;
  float s_pen = 0.f, s_con = 0.f, s_npts = 0.f, s_cons = 0.f, s_lo = 0.f;
  if (j < NO) {
    int o = b * NO + j;
    float oxp = obj[o * 3 + 0], oyp = obj[o * 3 + 1], ozp = obj[o * 3 + 2];
    float rd2 = recon_d2[o]; int ri = recon_idx[o];
    float gd2 = gt_d2[o];    int gi = gt_idx[o];
    float pd2 = prior_d2[o];

    const float* nr = &recon[(b * NH + ri) * 3];
    const float* nn = &rnorm[(b * NH + ri) * 3];
    float drx = oxp - nr[0], dry = oyp - nr[1], drz = ozp - nr[2];
    float dotr = nn[0] * drx + nn[1] * dry + nn[2] * drz;
    float o2h = sqrtf(rd2) * sgnf(dotr);
    bool interior = (-dotr) > 0.f;        // dot(nn_hand - obj, normal) > 0
    s_pen = interior ? rd2 : 0.f;

    const float* ng = &gtp[(b * NH + gi) * 3];
    const float* nm = &gnorm[(b * NH + gi) * 3];
    float dgx = oxp - ng[0], dgy = oyp - ng[1], dgz = ozp - ng[2];
    float dotg = nm[0] * dgx + nm[1] * dgy + nm[2] * dgz;
    float o2hg = sqrtf(gd2) * sgnf(dotg);

    bool cmap  = sqrtf(gd2) < 0.005f;
    bool rcmap = sqrtf(rd2) < 0.005f;
    s_con  = cmap ? pd2 : 0.f;
    s_npts = cmap ? 1.f : 0.f;
    s_cons = (cmap && rcmap) ? 1.f : 0.f;

    bool  w_dist = (o2hg < 0.01f) && (o2hg > -0.005f);
    bool  w_neg  = o2h < 0.f;
    float w = w_neg ? 1.5f : (w_dist ? 1.f : 0.1f);
    s_lo = fabsf(o2h - o2hg) * w;
  }
  s_pen = wave_sum(s_pen); s_con = wave_sum(s_con); s_npts = wave_sum(s_npts);
  s_cons = wave_sum(s_cons); s_lo = wave_sum(s_lo);
  if ((threadIdx.x & 31) == 0) {
    atomicAdd(&acc[0], s_pen);
    atomicAdd(&acc[1], s_con);
    atomicAdd(&acc[2], s_npts);
    atomicAdd(&acc[3], s_cons);
    atomicAdd(&acc[4], s_lo);
  }
}

// ---------------------------------------------------------------------------
// Per hand-point terms: loss_h and recon MSE.  acc[5]=sum_loss_h acc[6]=sum_rec
// ---------------------------------------------------------------------------
__global__ __launch_bounds__(256) void gcl_post_hand(
    const float* __restrict__ recon, const float* __restrict__ gtp,
    const float* __restrict__ vw,
    const unsigned* __restrict__ h2o_min, const unsigned* __restrict__ h2ogt_min,
    float* __restrict__ acc) {
  int t = blockIdx.x * blockDim.x + threadIdx.x;
  float s_lh = 0.f, s_rec = 0.f;
  if (t < NB * NH) {
    int i = t % NH;
    float h2o  = sqrtf(fmaxf(__uint_as_float(h2o_min[t]), 0.f));
    float h2og = sqrtf(fmaxf(__uint_as_float(h2ogt_min[t]), 0.f));
    float vw2  = powf(vw[i], 0.4f);   // v ** (1/2.5)
    s_lh = fabsf(h2o - h2og) * vw2;
    float dx = recon[t * 3 + 0] - gtp[t * 3 + 0];
    float dy = recon[t * 3 + 1] - gtp[t * 3 + 1];
    float dz = recon[t * 3 + 2] - gtp[t * 3 + 2];
    s_rec = dx * dx + dy * dy + dz * dz;
  }
  s_lh = wave_sum(s_lh); s_rec = wave_sum(s_rec);
  if ((threadIdx.x & 31) == 0) {
    atomicAdd(&acc[5], s_lh);
    atomicAdd(&acc[6], s_rec);
  }
}

// acc[7] = sum(1 + log_var - mean^2 - exp(log_var))
__global__ __launch_bounds__(256) void gcl_kld(
    const float* __restrict__ mean, const float* __restrict__ log_var,
    float* __restrict__ acc) {
  int t = blockIdx.x * blockDim.x + threadIdx.x;
  float s = 0.f;
  if (t < NB * NZ) {
    float m = mean[t], lv = log_var[t];
    s = 1.f + lv - m * m - expf(lv);
  }
  s = wave_sum(s);
  if ((threadIdx.x & 31) == 0) atomicAdd(&acc[7], s);
}

__global__ void gcl_final(const float* __restrict__ acc, float* __restrict__ out) {
  if (threadIdx.x == 0 && blockIdx.x == 0) {
    const float Bf = (float)NB;
    float recon_loss  = acc[6] / Bf;
    float kld         = -0.5f * acc[7] / Bf * 10.f;
    float penetr      = 100.f * acc[0] / Bf;
    float npts        = acc[2];
    float contact     = 3000.f * ((npts > 0.f) ? (acc[1] / (Bf * npts)) : 0.f);
    float consistency = -5.f * acc[3] / (npts + 0.0001f);
    float kc     = 1.f - 0.005f;  // 1 - KL_COEF
    float loss_h = 35.f * kc * (acc[5] / (Bf * (float)NH));
    float loss_o = 30.f * kc * (acc[4] / (Bf * (float)NO));
    out[0] = 1.0f * recon_loss + 0.1f * kld + 1000.f * penetr +
             10.f * contact + 10.f * consistency + loss_h + loss_o;
  }
}

// ---------------------------------------------------------------------------
extern "C" void kernel_launch(void* const* d_in, const int* in_sizes, int n_in,
                              void* d_out, int out_size, void* d_ws, size_t ws_size,
                              hipStream_t stream) {
  const float* recon_xyz     = (const float*)d_in[0];
  const float* gt_xyz        = (const float*)d_in[1];
  const float* recon_normals = (const float*)d_in[2];
  const float* gt_normals    = (const float*)d_in[3];
  const float* obj_xyz       = (const float*)d_in[4];
  const float* mean          = (const float*)d_in[5];
  const float* log_var       = (const float*)d_in[6];
  const float* v_weights     = (const float*)d_in[7];
  float* out = (float*)d_out;

  char* p = (char*)d_ws;
  auto carve = [&](size_t bytes) -> void* {
    void* r = (void*)p;
    p += (bytes + 255) & ~(size_t)255;
    return r;
  };
  float*    recon_d2  = (float*)carve(sizeof(float) * NB * NO);
  int*      recon_idx = (int*)carve(sizeof(int) * NB * NO);
  float*    gt_d2     = (float*)carve(sizeof(float) * NB * NO);
  int*      gt_idx    = (int*)carve(sizeof(int) * NB * NO);
  float*    prior_d2  = (float*)carve(sizeof(float) * NB * NO);
  unsigned* h2o_min   = (unsigned*)carve(sizeof(unsigned) * NB * NH);
  unsigned* h2ogt_min = (unsigned*)carve(sizeof(unsigned) * NB * NH);
  float*    acc       = (float*)carve(sizeof(float) * 16);

  {
    int n = NB * NH;
    gcl_init<<<(n + 255) / 256, 256, 0, stream>>>(h2o_min, h2ogt_min, acc);
  }
  {
    dim3 grid((NO + ROWS_PER_BLOCK - 1) / ROWS_PER_BLOCK, NB, 2);
    gcl_dist<<<grid, 256, 0, stream>>>(obj_xyz, recon_xyz, gt_xyz,
                                       recon_d2, recon_idx, gt_d2, gt_idx,
                                       prior_d2, h2o_min, h2ogt_min);
  }
  {
    dim3 grid((NO + 255) / 256, NB);
    gcl_post_obj<<<grid, 256, 0, stream>>>(obj_xyz, recon_xyz, recon_normals,
                                           gt_xyz, gt_normals,
                                           recon_d2, recon_idx, gt_d2, gt_idx,
                                           prior_d2, acc);
  }
  {
    int n = NB * NH;
    gcl_post_hand<<<(n + 255) / 256, 256, 0, stream>>>(
        recon_xyz, gt_xyz, v_weights, h2o_min, h2ogt_min, acc);
  }
  {
    int n = NB * NZ;
    gcl_kld<<<(n + 255) / 256, 256, 0, stream>>>(mean, log_var, acc);
  }
  gcl_final<<<1, 32, 0, stream>>>(acc, out);
}